// NGCF_60911226192491
// MI455X (gfx1250) — compile-verified
//
#include <hip/hip_runtime.h>
#include <hip/hip_bf16.h>

#define N_USERS 100000
#define N_ITEMS 200000
#define N_NODES (N_USERS + N_ITEMS)
#define EMB 64
#define NLAYERS 3
#define N_EDGES 9600000
#define BATCH 4096
#define NEG_SLOPE 0.2f

typedef __attribute__((ext_vector_type(2))) float v2f;
typedef __attribute__((ext_vector_type(4))) float v4f;
typedef __attribute__((ext_vector_type(8))) float v8f;

// ---------------------------------------------------------------------------
// ego0 = concat(user_emb, item_emb)  (float4 copies, exact sizing)
// ---------------------------------------------------------------------------
__global__ __launch_bounds__(256) void ngcf_concat_init(
    const float* __restrict__ ue, const float* __restrict__ ie,
    float* __restrict__ ego)
{
    size_t f = ((size_t)blockIdx.x * 256 + threadIdx.x) * 4;
    const size_t UFL = (size_t)N_USERS * EMB;
    if (f < UFL) {
        *(v4f*)(ego + f) = *(const v4f*)(ue + f);
    } else {
        *(v4f*)(ego + f) = *(const v4f*)(ie + (f - UFL));
    }
}

// ---------------------------------------------------------------------------
// zero the neigh accumulator
// ---------------------------------------------------------------------------
__global__ __launch_bounds__(256) void ngcf_zero(float* __restrict__ p)
{
    size_t f = ((size_t)blockIdx.x * 256 + threadIdx.x) * 4;
    v4f z = {0.f, 0.f, 0.f, 0.f};
    *(v4f*)(p + f) = z;
}

// ---------------------------------------------------------------------------
// Edge scatter: one wave per edge; lane handles 2 of the 64 dims.
// neigh[row] += ego[col] * val   (f32 atomics; neigh fits in 192MB L2)
// ---------------------------------------------------------------------------
__global__ __launch_bounds__(256) void ngcf_scatter(
    const int* __restrict__ erow, const int* __restrict__ ecol,
    const float* __restrict__ eval, const float* __restrict__ ego,
    float* __restrict__ neigh)
{
    unsigned gid = blockIdx.x * 256u + threadIdx.x;
    unsigned e = gid >> 5;
    if (e >= N_EDGES) return;
    int lane = threadIdx.x & 31;
    int row = erow[e];
    int col = ecol[e];
    float v = eval[e];
    v2f g = *(const v2f*)(ego + (size_t)col * EMB + lane * 2);
    float* dst = neigh + (size_t)row * EMB + lane * 2;
    atomicAdd(dst + 0, g.x * v);
    atomicAdd(dst + 1, g.y * v);
}

// ---------------------------------------------------------------------------
// Fused layer epilogue, one wave per 16-node tile:
//   D = neigh@Wgc + (ego*neigh)@Wbi + (bgc+bbi)    via V_WMMA_F32_16X16X4_F32
//   ego_out = l2norm(leaky_relu(D))                via LDS tile + shfl reduce
// ---------------------------------------------------------------------------
__global__ __launch_bounds__(256) void ngcf_gemm_norm(
    const float* __restrict__ ego, const float* __restrict__ neigh,
    const float* __restrict__ Wgc, const float* __restrict__ bgc,
    const float* __restrict__ Wbi, const float* __restrict__ bbi,
    float* __restrict__ ego_out)
{
    __shared__ float tile[8][16 * EMB];          // 4KB per wave, 32KB total

    const int lane = threadIdx.x & 31;
    const int wv   = threadIdx.x >> 5;
    const int tid  = blockIdx.x * 8 + wv;        // 16-row tile id (wave-uniform)
    if (tid >= N_NODES / 16) return;             // wave-uniform exit: EXEC all-1 for WMMA
    const int row0 = tid * 16;

    const int m    = lane & 15;                  // A row within tile
    const int koff = (lane >> 4) * 2;            // 0 (lanes 0-15) or 2 (lanes 16-31)
    const int ncol = lane & 15;                  // B/C/D column within N-tile

    // Preload A operands for all 16 K-steps (K=64): neigh row and ego*neigh row
    v2f An[16], Ab[16];
#pragma unroll
    for (int ks = 0; ks < 16; ++ks) {
        size_t base = (size_t)(row0 + m) * EMB + 4 * ks + koff;
        v2f n2 = *(const v2f*)(neigh + base);
        v2f e2 = *(const v2f*)(ego + base);
        An[ks] = n2;
        Ab[ks] = n2 * e2;                        // bilinear term (ego .* neigh)
    }

#pragma unroll
    for (int t = 0; t < 4; ++t) {                // four 16-col N-tiles of the 64 cols
        const int c = 16 * t + ncol;
        float bias = bgc[c] + bbi[c];
        v8f acc;
#pragma unroll
        for (int i = 0; i < 8; ++i) acc[i] = bias;   // col is lane-invariant per component

#pragma unroll
        for (int ks = 0; ks < 16; ++ks) {
            int kr = 4 * ks + koff;
            v2f bg = { Wgc[(size_t)(kr + 0) * EMB + c], Wgc[(size_t)(kr + 1) * EMB + c] };
            v2f bb = { Wbi[(size_t)(kr + 0) * EMB + c], Wbi[(size_t)(kr + 1) * EMB + c] };
            acc = __builtin_amdgcn_wmma_f32_16x16x4_f32(
                false, An[ks], false, bg, (short)0, acc, false, false);
            acc = __builtin_amdgcn_wmma_f32_16x16x4_f32(
                false, Ab[ks], false, bb, (short)0, acc, false, false);
        }

        // leaky relu, park tile in this wave's LDS slice (D layout -> row major)
#pragma unroll
        for (int i = 0; i < 8; ++i) {
            float x = acc[i];
            x = x > 0.f ? x : NEG_SLOPE * x;
            int r = (lane >> 4) * 8 + i;
            tile[wv][r * EMB + c] = x;
        }
    }

    // Row L2 norm: lane pair (l, l+16) handles row l&15, each sums half the row.
    // Same-wave LDS RAW is in-order (DS ops ordered per wave).
    {
        int r = lane & 15;
        int cbase = (lane >> 4) * 32;
        const float* rowp = &tile[wv][r * EMB];
        float s = 0.f;
#pragma unroll
        for (int cc = 0; cc < 32; ++cc) {
            float x = rowp[cbase + cc];
            s += x * x;
        }
        s += __shfl_xor(s, 16, 32);
        float inv = 1.f / fmaxf(sqrtf(s), 1e-12f);
        float* op = ego_out + (size_t)(row0 + r) * EMB + cbase;
#pragma unroll
        for (int cc = 0; cc < 32; ++cc) op[cc] = rowp[cbase + cc] * inv;
    }
}

// ---------------------------------------------------------------------------
// Final batch gather: one wave per (set, b); 256-col concat row from 4 egos.
// out layout: [user_all[users] | item_all[pos] | item_all[neg]] flat
// ---------------------------------------------------------------------------
__global__ __launch_bounds__(256) void ngcf_gather(
    const int* __restrict__ users, const int* __restrict__ pos,
    const int* __restrict__ neg,
    const float* __restrict__ e0, const float* __restrict__ e1,
    const float* __restrict__ e2, const float* __restrict__ e3,
    float* __restrict__ out)
{
    unsigned w = (blockIdx.x * 256u + threadIdx.x) >> 5;
    if (w >= 3u * BATCH) return;
    int lane = threadIdx.x & 31;
    int set = w / BATCH;
    int b = w % BATCH;
    int node = (set == 0) ? users[b]
             : (set == 1) ? (N_USERS + pos[b])
                          : (N_USERS + neg[b]);
    float* o = out + (size_t)w * (4 * EMB);
    const float* srcs[4] = {e0, e1, e2, e3};
#pragma unroll
    for (int l = 0; l < 4; ++l) {
        v2f x = *(const v2f*)(srcs[l] + (size_t)node * EMB + lane * 2);
        *(v2f*)(o + l * EMB + lane * 2) = x;
    }
}

// ---------------------------------------------------------------------------
extern "C" void kernel_launch(void* const* d_in, const int* in_sizes, int n_in,
                              void* d_out, int out_size, void* d_ws, size_t ws_size,
                              hipStream_t stream)
{
    const int*   erow = (const int*)  d_in[0];
    const int*   ecol = (const int*)  d_in[1];
    const float* eval = (const float*)d_in[2];
    const float* uemb = (const float*)d_in[3];
    const float* iemb = (const float*)d_in[4];
    const float* Wgc  = (const float*)d_in[5];   // [3,64,64]
    const float* bgc  = (const float*)d_in[6];   // [3,64]
    const float* Wbi  = (const float*)d_in[7];
    const float* bbi  = (const float*)d_in[8];
    const int*   users = (const int*) d_in[9];
    const int*   pos   = (const int*) d_in[10];
    const int*   neg   = (const int*) d_in[11];
    float* out = (float*)d_out;

    const size_t NN64 = (size_t)N_NODES * EMB;   // 19.2M floats
    float* ws = (float*)d_ws;
    float* ego[4] = { ws, ws + NN64, ws + 2 * NN64, ws + 3 * NN64 };
    float* neigh = ws + 4 * NN64;

    const int nodeVecBlocks = (int)(NN64 / 4 / 256);      // 18750, exact
    ngcf_concat_init<<<nodeVecBlocks, 256, 0, stream>>>(uemb, iemb, ego[0]);

    const int scatterBlocks = (int)(((size_t)N_EDGES * 32) / 256);  // 1.2M, exact
    const int gemmBlocks = (N_NODES / 16 + 7) / 8;        // 2344

    for (int k = 0; k < NLAYERS; ++k) {
        ngcf_zero<<<nodeVecBlocks, 256, 0, stream>>>(neigh);
        ngcf_scatter<<<scatterBlocks, 256, 0, stream>>>(erow, ecol, eval, ego[k], neigh);
        ngcf_gemm_norm<<<gemmBlocks, 256, 0, stream>>>(
            ego[k], neigh,
            Wgc + (size_t)k * EMB * EMB, bgc + (size_t)k * EMB,
            Wbi + (size_t)k * EMB * EMB, bbi + (size_t)k * EMB,
            ego[k + 1]);
    }

    const int gatherBlocks = (3 * BATCH * 32) / 256;      // 1536, exact
    ngcf_gather<<<gatherBlocks, 256, 0, stream>>>(
        users, pos, neg, ego[0], ego[1], ego[2], ego[3], out);
}